// Place_Model_62148176773170
// MI455X (gfx1250) — compile-verified
//
#include <hip/hip_runtime.h>

// ---------------------------------------------------------------------------
// PointNet++ place model, CDNA5 (gfx1250).
// All GEMM work via V_WMMA_F32_16X16X4_F32, 2x4 output tiles per wave32
// (8 wmma per K-step, zero predication: K padded to %4 via ldx + packed W,
// M%32==0 and N%64==0 hold for every layer in this model).
// ---------------------------------------------------------------------------

#ifndef __has_builtin
#define __has_builtin(x) 0
#endif
#if __has_builtin(__builtin_amdgcn_wmma_f32_16x16x4_f32)
#define HAS_WMMA_F32 1
#else
#define HAS_WMMA_F32 0
#endif

typedef __attribute__((ext_vector_type(2))) float v2f;
typedef __attribute__((ext_vector_type(8))) float v8f;

#define BATCH 8
#define NPTS  4096
#define BN_SCALE_F 0.99999500003749971894f /* 1/sqrt(1+1e-5) */

static inline int nblk(long n, int t) { return (int)((n + t - 1) / t); }
static inline int pad4(int k) { return (k + 3) & ~3; }

// ---------------------------------------------------------------------------
// (B,3,N) -> (B,N,3)
__global__ void k_transpose(const float* __restrict__ in, float* __restrict__ out,
                            int Bn, int Nn) {
  int t = blockIdx.x * blockDim.x + threadIdx.x;
  if (t >= Bn * Nn) return;
  int b = t / Nn, n = t - b * Nn;
#pragma unroll
  for (int d = 0; d < 3; ++d)
    out[(size_t)t * 3 + d] = in[((size_t)b * 3 + d) * Nn + n];
}

// (B,N,3) -> (B,N,6) : concat([xyz, xyz]) (l0_points == l0_xyz)
__global__ void k_dup6(const float* __restrict__ xyz, float* __restrict__ out, long M) {
  long t = blockIdx.x * (long)blockDim.x + threadIdx.x;
  if (t >= M) return;
  float x = xyz[t * 3], y = xyz[t * 3 + 1], z = xyz[t * 3 + 2];
  float* o = out + t * 6;
  o[0] = x; o[1] = y; o[2] = z; o[3] = x; o[4] = y; o[5] = z;
}

// ---------------------------------------------------------------------------
// Farthest point sampling: one block per batch, dist[] resident in LDS.
// Matches jnp: first index 0; argmax prefers lowest index on tie.
__global__ void k_fps(const float* __restrict__ xyz, int* __restrict__ out,
                      int Nn, int npoint) {
  __shared__ float dist[4096];
  __shared__ float rv[256];
  __shared__ int   ri[256];
  __shared__ int   s_far;
  const int b   = blockIdx.x;
  const int tid = threadIdx.x;
  const float* p = xyz + (size_t)b * Nn * 3;
  for (int i = tid; i < Nn; i += blockDim.x) dist[i] = 1e10f;
  if (tid == 0) s_far = 0;
  __syncthreads();
  for (int it = 0; it < npoint; ++it) {
    const int far = s_far;
    if (tid == 0) out[b * npoint + it] = far;
    const float cx = p[far * 3 + 0], cy = p[far * 3 + 1], cz = p[far * 3 + 2];
    float best = -1.0f; int bi = Nn - 1;
    for (int i = tid; i < Nn; i += blockDim.x) {
      float dx = p[i * 3 + 0] - cx, dy = p[i * 3 + 1] - cy, dz = p[i * 3 + 2] - cz;
      float d  = dx * dx + dy * dy + dz * dz;
      float nd = fminf(dist[i], d);
      dist[i]  = nd;
      if (nd > best) { best = nd; bi = i; }
    }
    rv[tid] = best; ri[tid] = bi;
    __syncthreads();
    for (int s = blockDim.x >> 1; s > 0; s >>= 1) {
      if (tid < s) {
        if (rv[tid + s] > rv[tid] ||
            (rv[tid + s] == rv[tid] && ri[tid + s] < ri[tid])) {
          rv[tid] = rv[tid + s]; ri[tid] = ri[tid + s];
        }
      }
      __syncthreads();
    }
    if (tid == 0) s_far = ri[0];
    __syncthreads();
  }
}

__global__ void k_gather3(const float* __restrict__ xyz, const int* __restrict__ idx,
                          float* __restrict__ out, int Nn, int S, int total) {
  int t = blockIdx.x * blockDim.x + threadIdx.x;
  if (t >= total) return;
  int b = t / S;
  int j = idx[t];
  const float* src = xyz + ((size_t)b * Nn + j) * 3;
  float* o = out + (size_t)t * 3;
  o[0] = src[0]; o[1] = src[1]; o[2] = src[2];
}

// ball query: first nsample in-radius indices ascending; pad with first found.
__global__ void k_ballquery(const float* __restrict__ xyz, const float* __restrict__ nxyz,
                            int* __restrict__ idx, int Nn, int S, int Kk, float r2,
                            int total) {
  int t = blockIdx.x * blockDim.x + threadIdx.x;
  if (t >= total) return;
  int b = t / S;
  const float* q = nxyz + (size_t)t * 3;
  const float* p = xyz + (size_t)b * Nn * 3;
  float qx = q[0], qy = q[1], qz = q[2];
  int* o = idx + (size_t)t * Kk;
  int cnt = 0, first = -1;
  for (int n = 0; n < Nn && cnt < Kk; ++n) {
    float dx = p[n * 3] - qx, dy = p[n * 3 + 1] - qy, dz = p[n * 3 + 2] - qz;
    float d = dx * dx + dy * dy + dz * dz;
    if (d <= r2) { if (first < 0) first = n; o[cnt++] = n; }
  }
  if (first < 0) first = Nn - 1;
  for (; cnt < Kk; ++cnt) o[cnt] = first;
}

// grouped rows (stride ldx, zero-padded): [xyz[idx]-new_xyz (3), points[idx] (Cp)]
__global__ void k_group(const float* __restrict__ xyz, const float* __restrict__ nxyz,
                        const float* __restrict__ pts, const int* __restrict__ idx,
                        float* __restrict__ out, int Nn, int S, int Kk, int Cp,
                        int ldx, long total) {
  long t = blockIdx.x * (long)blockDim.x + threadIdx.x;
  if (t >= total) return;
  long bs = t / Kk;
  int  b  = (int)(bs / S);
  int  j  = idx[t];
  const float* q  = nxyz + (size_t)bs * 3;
  const float* pj = xyz + ((size_t)b * Nn + j) * 3;
  float* o = out + t * (size_t)ldx;
  o[0] = pj[0] - q[0]; o[1] = pj[1] - q[1]; o[2] = pj[2] - q[2];
  const float* fj = pts + ((size_t)b * Nn + j) * Cp;
  for (int c = 0; c < Cp; ++c) o[3 + c] = fj[c];
  for (int c = 3 + Cp; c < ldx; ++c) o[c] = 0.f;
}

// max over K: x (BS, K, C) -> out (BS, C)
__global__ void k_maxpool(const float* __restrict__ x, float* __restrict__ out,
                          long BS, int K, int C) {
  long t = blockIdx.x * (long)blockDim.x + threadIdx.x;
  if (t >= BS * C) return;
  long bs = t / C; int c = (int)(t - bs * C);
  const float* p = x + bs * (size_t)K * C + c;
  float m = -3.4e38f;
  for (int k = 0; k < K; ++k) m = fmaxf(m, p[(size_t)k * C]);
  out[t] = m;
}

// concat with optional broadcast source; output stride ldx, zero pad tail.
__global__ void k_concat(const float* __restrict__ A, int Ca,
                         const float* __restrict__ Bs, int Cb, long bStride,
                         int rowsPerB, float* __restrict__ out, long M, int ldx,
                         int bcFirst) {
  long t = blockIdx.x * (long)blockDim.x + threadIdx.x;
  if (t >= M * ldx) return;
  long m = t / ldx; int c = (int)(t - m * ldx);
  int C = Ca + Cb;
  float v = 0.f;
  if (c < C) {
    if (!bcFirst) {
      v = (c < Ca) ? A[m * (size_t)Ca + c]
                   : Bs[(size_t)(m / rowsPerB) * bStride + (c - Ca)];
    } else {
      v = (c < Cb) ? Bs[(size_t)(m / rowsPerB) * bStride + c]
                   : A[m * (size_t)Ca + (c - Cb)];
    }
  }
  out[t] = v;
}

// 3-NN inverse-distance interp fused with concat (output stride ldx, zero pad).
__global__ void k_three_interp(const float* __restrict__ xyz1, const float* __restrict__ xyz2,
                               const float* __restrict__ p1, int C1,
                               const float* __restrict__ p2, int C2,
                               float* __restrict__ out, int Bb, int Nn, int S, int ldx) {
  int t = blockIdx.x * blockDim.x + threadIdx.x;
  if (t >= Bb * Nn) return;
  int b = t / Nn;
  const float* q  = xyz1 + (size_t)t * 3;
  const float* xs = xyz2 + (size_t)b * S * 3;
  float qx = q[0], qy = q[1], qz = q[2];
  float d0 = 3.4e38f, d1 = 3.4e38f, d2 = 3.4e38f;
  int   i0 = 0, i1 = 0, i2 = 0;
  for (int s = 0; s < S; ++s) {
    float dx = qx - xs[s * 3], dy = qy - xs[s * 3 + 1], dz = qz - xs[s * 3 + 2];
    float d = dx * dx + dy * dy + dz * dz;
    if (d < d0)      { d2 = d1; i2 = i1; d1 = d0; i1 = i0; d0 = d; i0 = s; }
    else if (d < d1) { d2 = d1; i2 = i1; d1 = d; i1 = s; }
    else if (d < d2) { d2 = d; i2 = s; }
  }
  float w0 = 1.0f / (d0 + 1e-8f), w1 = 1.0f / (d1 + 1e-8f), w2 = 1.0f / (d2 + 1e-8f);
  float ws = w0 + w1 + w2;
  w0 /= ws; w1 /= ws; w2 /= ws;
  float* o = out + (size_t)t * ldx;
  const float* r1 = p1 + (size_t)t * C1;
  for (int c = 0; c < C1; ++c) o[c] = r1[c];
  const float* b0 = p2 + ((size_t)b * S + i0) * C2;
  const float* b1 = p2 + ((size_t)b * S + i1) * C2;
  const float* b2 = p2 + ((size_t)b * S + i2) * C2;
  for (int c = 0; c < C2; ++c)
    o[C1 + c] = w0 * b0[c] + w1 * b1[c] + w2 * b2[c];
  for (int c = C1 + C2; c < ldx; ++c) o[c] = 0.f;
}

// pack ragged-K weight into zero-padded (K4 x N) copy
__global__ void k_packW(const float* __restrict__ W, float* __restrict__ Wp,
                        int K, int K4, int N) {
  long t = blockIdx.x * (long)blockDim.x + threadIdx.x;
  if (t >= (long)K4 * N) return;
  int k = (int)(t / N), n = (int)(t - (long)k * N);
  Wp[t] = (k < K) ? W[(size_t)k * N + n] : 0.f;
}

// ---------------------------------------------------------------------------
// Fused Y = relu((X[M,K] @ W[K,N] + bias) * (BN_SCALE*gamma) + beta)
// Preconditions (hold for every layer in this model): M%32==0, N%64==0, K%4==0,
// X leading dim ldx (>=K, %4==0, pad cols zero), W is (K x N) dense.
// Each wave32 computes a 32x64 block: 2 (M) x 4 (N) WMMA tiles, 8 wmma/K-step.
// All loop addressing via pointer increments (constant strides) so the hot
// loop is just: 2x global_load_b64 (A) + 8x global_load_b32 (B) + 8x wmma.
__global__ void k_linear(const float* __restrict__ X, int ldx,
                         const float* __restrict__ W,
                         const float* __restrict__ bias, const float* __restrict__ gamma,
                         const float* __restrict__ beta, float* __restrict__ Y,
                         int M, int K, int N) {
  const int lane    = threadIdx.x & 31;
  const int wave    = threadIdx.x >> 5;
  const int tilesN4 = N >> 6;                 // groups of 4 N-tiles
  const int tilesM2 = M >> 5;                 // groups of 2 M-tiles
  const int tile    = blockIdx.x * (blockDim.x >> 5) + wave;
  if (tile >= tilesM2 * tilesN4) return;      // wave-uniform: EXEC stays all-ones
  const int mt   = tile / tilesN4;
  const int nt   = tile - mt * tilesN4;
  const int half = lane >> 4;
  const int l15  = lane & 15;
  const int row0 = mt * 32 + l15;             // A row, tile i adds i*16
  const int col0 = nt * 64 + l15;             // B col, tile j adds j*16

  v8f acc[2][4];
#pragma unroll
  for (int i = 0; i < 2; ++i)
#pragma unroll
    for (int j = 0; j < 4; ++j)
      acc[i][j] = (v8f){0.f, 0.f, 0.f, 0.f, 0.f, 0.f, 0.f, 0.f};

  // hoisted bases; advanced by constant strides inside the loop
  const float* xr0 = X + (size_t)row0 * ldx + half * 2;          // A rows (tile 0)
  const float* xr1 = xr0 + (size_t)16 * ldx;                     // A rows (tile 1)
  const float* wr0 = W + (size_t)(half * 2) * N + col0;          // B row ka
  const float* wr1 = wr0 + N;                                    // B row ka+1
  const size_t wStep = (size_t)4 * N;

  for (int k0 = 0; k0 < K; k0 += 4) {
#if HAS_WMMA_F32
    v2f a0 = *(const v2f*)xr0;                // 8B-aligned: ldx%4==0, half*2 even
    v2f a1 = *(const v2f*)xr1;
#pragma unroll
    for (int j = 0; j < 4; ++j) {
      v2f b;
      b.x = wr0[16 * j];
      b.y = wr1[16 * j];
      acc[0][j] = __builtin_amdgcn_wmma_f32_16x16x4_f32(
          false, a0, false, b, (short)0, acc[0][j], false, false);
      acc[1][j] = __builtin_amdgcn_wmma_f32_16x16x4_f32(
          false, a1, false, b, (short)0, acc[1][j], false, false);
    }
#else
    // scalar fallback (host pass / builtin missing), same tile mapping
#pragma unroll
    for (int i = 0; i < 2; ++i)
#pragma unroll
      for (int j = 0; j < 4; ++j)
#pragma unroll
        for (int r = 0; r < 8; ++r) {
          int row = mt * 32 + i * 16 + half * 8 + r;
          int col = col0 + 16 * j;
          float s = 0.f;
          for (int kk = k0; kk < k0 + 4; ++kk)
            s += X[(size_t)row * ldx + kk] * W[(size_t)kk * N + col];
          acc[i][j][r] += s;
        }
#endif
    xr0 += 4; xr1 += 4;
    wr0 += wStep; wr1 += wStep;
  }

  const int rowBase = mt * 32 + half * 8;
#pragma unroll
  for (int j = 0; j < 4; ++j) {
    const int col = col0 + 16 * j;
    const float sc = BN_SCALE_F * gamma[col];
    const float sh = beta[col];
    const float bi = bias[col];
#pragma unroll
    for (int i = 0; i < 2; ++i) {
#pragma unroll
      for (int r = 0; r < 8; ++r) {
        const int row = rowBase + i * 16 + r;
        float v = (acc[i][j][r] + bi) * sc + sh;
        Y[(size_t)row * N + col] = fmaxf(v, 0.f);
      }
    }
  }
}

// head conv2 (K->1) + sigmoid
__global__ void k_head(const float* __restrict__ X, const float* __restrict__ W,
                       const float* __restrict__ b, float* __restrict__ out,
                       long M, int K) {
  long t = blockIdx.x * (long)blockDim.x + threadIdx.x;
  if (t >= M) return;
  float acc = b[0];
  const float* x = X + (size_t)t * K;
  for (int k = 0; k < K; ++k) acc += x[k] * W[k];
  out[t] = 1.0f / (1.0f + expf(-acc));
}

// ---------------------------------------------------------------------------
// Host-side driver
// ---------------------------------------------------------------------------
struct Lyr { const float *W, *b, *g, *be; int cin, cout; };
struct Branch { Lyr sa1[3], sa2[3], sa3[3], fp3[2], fp2[2], fp1[3]; };

static const int SA1[4] = {6, 64, 64, 128};
static const int SA2[4] = {131, 128, 128, 256};
static const int SA3[4] = {259, 256, 512, 1024};
static const int FP3[3] = {1280, 256, 256};
static const int FP2[3] = {384, 256, 128};
static const int FP1[4] = {134, 128, 128, 128};

static int fill_group(void* const* d_in, int idx, Lyr* out, const int* ch, int nl) {
  int cin = ch[0];
  for (int i = 0; i < nl; ++i) {
    out[i].W  = (const float*)d_in[idx++];
    out[i].b  = (const float*)d_in[idx++];
    out[i].g  = (const float*)d_in[idx++];
    out[i].be = (const float*)d_in[idx++];
    out[i].cin = cin; out[i].cout = ch[i + 1]; cin = ch[i + 1];
  }
  return idx;
}

static int fill_branch(void* const* d_in, int idx, Branch& Br, bool sorted) {
  if (!sorted) {  // dict insertion order: sa1, sa2, sa3, fp3, fp2, fp1
    idx = fill_group(d_in, idx, Br.sa1, SA1, 3);
    idx = fill_group(d_in, idx, Br.sa2, SA2, 3);
    idx = fill_group(d_in, idx, Br.sa3, SA3, 3);
    idx = fill_group(d_in, idx, Br.fp3, FP3, 2);
    idx = fill_group(d_in, idx, Br.fp2, FP2, 2);
    idx = fill_group(d_in, idx, Br.fp1, FP1, 3);
  } else {        // jax-sorted keys: fp1, fp2, fp3, sa1, sa2, sa3
    idx = fill_group(d_in, idx, Br.fp1, FP1, 3);
    idx = fill_group(d_in, idx, Br.fp2, FP2, 2);
    idx = fill_group(d_in, idx, Br.fp3, FP3, 2);
    idx = fill_group(d_in, idx, Br.sa1, SA1, 3);
    idx = fill_group(d_in, idx, Br.sa2, SA2, 3);
    idx = fill_group(d_in, idx, Br.sa3, SA3, 3);
  }
  return idx;
}

// replace ragged-K weights with zero-padded packed copies in the arena
static void pack_layer(Lyr& L, float*& p, hipStream_t st) {
  if ((L.cin & 3) == 0) return;
  const int K4 = pad4(L.cin);
  const long tot = (long)K4 * L.cout;
  k_packW<<<nblk(tot, 256), 256, 0, st>>>(L.W, p, L.cin, K4, L.cout);
  L.W = p; L.cin = K4;
  p += tot;
}
static void pack_branch(Branch& Br, float*& p, hipStream_t st) {
  for (int i = 0; i < 3; ++i) pack_layer(Br.sa1[i], p, st);
  for (int i = 0; i < 3; ++i) pack_layer(Br.sa2[i], p, st);
  for (int i = 0; i < 3; ++i) pack_layer(Br.sa3[i], p, st);
  for (int i = 0; i < 2; ++i) pack_layer(Br.fp3[i], p, st);
  for (int i = 0; i < 2; ++i) pack_layer(Br.fp2[i], p, st);
  for (int i = 0; i < 3; ++i) pack_layer(Br.fp1[i], p, st);
}

static void launch_linear(const float* X, int ldx, const Lyr& L, float* Y, long M,
                          hipStream_t st) {
  long tiles = (M >> 5) * ((long)L.cout >> 6);   // 32x64 blocks per wave
  k_linear<<<nblk(tiles, 8), 256, 0, st>>>(X, ldx, L.W, L.b, L.g, L.be, Y,
                                           (int)M, L.cin, L.cout);
}

static const float* run_mlp(const float* X0, int ldx0, const Lyr* Ls, int nl, long M,
                            float* ping, float* pong, float* final_out, hipStream_t st) {
  const float* cur = X0;
  int ldx = ldx0;
  for (int i = 0; i < nl; ++i) {
    float* dst = (i == nl - 1) ? final_out : ((i & 1) ? pong : ping);
    launch_linear(cur, ldx, Ls[i], dst, M, st);
    cur = dst;
    ldx = Ls[i].cout;
  }
  return cur;
}

struct Ctx {
  float *l0_xyz, *l0_cat, *l1_xyz, *l1_p, *l2_xyz, *l2_p, *l3_p, *l2_p2, *l1_p2;
  float *Xbuf, *ping, *pong;
  int *fps1, *bq1, *fps2, *bq2;
};

static void sa_stage(const float* in_xyz, const float* in_pts, int Nin, int S, int Kk,
                     float radius, int Cp, const Lyr* Ls, int nl, int* fpsIdx, int* bqIdx,
                     float* new_xyz, float* out_p, const Ctx& C, hipStream_t st) {
  k_fps<<<BATCH, 256, 0, st>>>(in_xyz, fpsIdx, Nin, S);
  const int bs = BATCH * S;
  k_gather3<<<nblk(bs, 256), 256, 0, st>>>(in_xyz, fpsIdx, new_xyz, Nin, S, bs);
  k_ballquery<<<nblk(bs, 256), 256, 0, st>>>(in_xyz, new_xyz, bqIdx, Nin, S, Kk,
                                             radius * radius, bs);
  const long Mrows = (long)bs * Kk;
  const int ldx = pad4(3 + Cp);
  k_group<<<nblk(Mrows, 256), 256, 0, st>>>(in_xyz, new_xyz, in_pts, bqIdx, C.Xbuf,
                                            Nin, S, Kk, Cp, ldx, Mrows);
  const float* last = run_mlp(C.Xbuf, ldx, Ls, nl, Mrows, C.ping, C.pong, C.ping, st);
  const int Cout = Ls[nl - 1].cout;
  k_maxpool<<<nblk((long)bs * Cout, 256), 256, 0, st>>>(last, out_p, (long)bs, Kk, Cout);
}

static void run_branch(const float* xyz_b3n, const Branch& Br, const Ctx& C,
                       float* f_out, hipStream_t st) {
  const int bn = BATCH * NPTS;
  k_transpose<<<nblk(bn, 256), 256, 0, st>>>(xyz_b3n, C.l0_xyz, BATCH, NPTS);
  k_dup6<<<nblk(bn, 256), 256, 0, st>>>(C.l0_xyz, C.l0_cat, (long)bn);
  // SA1: 4096 -> 512 centers, 32 nbrs, r=0.2 -> MLP 6(->8)->64->64->128
  sa_stage(C.l0_xyz, C.l0_xyz, NPTS, 512, 32, 0.2f, 3, Br.sa1, 3,
           C.fps1, C.bq1, C.l1_xyz, C.l1_p, C, st);
  // SA2: 512 -> 128 centers, 64 nbrs, r=0.4 -> MLP 131(->132)->128->128->256
  sa_stage(C.l1_xyz, C.l1_p, 512, 128, 64, 0.4f, 128, Br.sa2, 3,
           C.fps2, C.bq2, C.l2_xyz, C.l2_p, C, st);
  // SA3 (group-all): rows = (B*128, 259->260) = [l2_xyz, l2_p]
  const long M3 = (long)BATCH * 128;
  k_concat<<<nblk(M3 * 260, 256), 256, 0, st>>>(C.l2_xyz, 3, C.l2_p, 256, 256, 1,
                                                C.Xbuf, M3, 260, 0);
  const float* last3 = run_mlp(C.Xbuf, 260, Br.sa3, 3, M3, C.ping, C.pong, C.ping, st);
  k_maxpool<<<nblk((long)BATCH * 1024, 256), 256, 0, st>>>(last3, C.l3_p, BATCH, 128, 1024);
  // FP3 (S==1 broadcast): x = [l2_p (256), l3_p bcast (1024)], ldx 1280
  k_concat<<<nblk(M3 * 1280, 256), 256, 0, st>>>(C.l2_p, 256, C.l3_p, 1024, 1024, 128,
                                                 C.Xbuf, M3, 1280, 0);
  run_mlp(C.Xbuf, 1280, Br.fp3, 2, M3, C.ping, C.pong, C.l2_p2, st);
  // FP2: 3-NN 128 -> 512, x = [l1_p (128), interp (256)], ldx 384
  const int bs2 = BATCH * 512;
  k_three_interp<<<nblk(bs2, 256), 256, 0, st>>>(C.l1_xyz, C.l2_xyz, C.l1_p, 128,
                                                 C.l2_p2, 256, C.Xbuf, BATCH, 512, 128, 384);
  run_mlp(C.Xbuf, 384, Br.fp2, 2, bs2, C.ping, C.pong, C.l1_p2, st);
  // FP1: 3-NN 512 -> 4096, x = [l0_cat (6), interp (128)], ldx 134->136
  k_three_interp<<<nblk(bn, 256), 256, 0, st>>>(C.l0_xyz, C.l1_xyz, C.l0_cat, 6,
                                                C.l1_p2, 128, C.Xbuf, BATCH, NPTS, 512, 136);
  run_mlp(C.Xbuf, 136, Br.fp1, 3, bn, C.ping, C.pong, f_out, st);
}

extern "C" void kernel_launch(void* const* d_in, const int* in_sizes, int n_in,
                              void* d_out, int out_size, void* d_ws, size_t ws_size,
                              hipStream_t stream) {
  (void)n_in; (void)out_size; (void)ws_size;

  // ----- input mapping (auto-detect harness flattening order) -----
  const float *xyzA = nullptr, *xyzB = nullptr;
  const float *c1W = nullptr, *c1b = nullptr, *g1 = nullptr, *be1 = nullptr,
              *c2W = nullptr, *c2b = nullptr;
  Branch pick{}, place{};
  if (in_sizes[0] == BATCH * 3 * NPTS) {
    xyzA = (const float*)d_in[0];
    xyzB = (const float*)d_in[1];
    if (in_sizes[2] == 6 * 64) {
      // full insertion order: pick, place, conv1_W, conv1_b, bn1_g, bn1_b, conv2_W, conv2_b
      int i = 2;
      i = fill_branch(d_in, i, pick, false);
      i = fill_branch(d_in, i, place, false);
      c1W = (const float*)d_in[i++]; c1b = (const float*)d_in[i++];
      g1  = (const float*)d_in[i++]; be1 = (const float*)d_in[i++];
      c2W = (const float*)d_in[i++]; c2b = (const float*)d_in[i++];
    } else {
      // params jax-sorted: bn1_b, bn1_g, conv1_W, conv1_b, conv2_W, conv2_b, pick, place
      int i = 2;
      be1 = (const float*)d_in[i++]; g1  = (const float*)d_in[i++];
      c1W = (const float*)d_in[i++]; c1b = (const float*)d_in[i++];
      c2W = (const float*)d_in[i++]; c2b = (const float*)d_in[i++];
      i = fill_branch(d_in, i, pick, true);
      i = fill_branch(d_in, i, place, true);
    }
  } else {
    // fully sorted top-level: params..., xyz_pick, xyz_place
    int i = 0;
    be1 = (const float*)d_in[i++]; g1  = (const float*)d_in[i++];
    c1W = (const float*)d_in[i++]; c1b = (const float*)d_in[i++];
    c2W = (const float*)d_in[i++]; c2b = (const float*)d_in[i++];
    i = fill_branch(d_in, i, pick, true);
    i = fill_branch(d_in, i, place, true);
    xyzA = (const float*)d_in[i++];
    xyzB = (const float*)d_in[i++];
  }

  // ----- workspace arena (floats) -----
  float* base = (float*)d_ws;
  size_t off = 0;
  auto alloc = [&](size_t n) { float* p = base + off; off += n; return p; };
  Ctx C{};
  C.l0_xyz = alloc(98304);     // B*4096*3
  C.l0_cat = alloc(196608);    // B*4096*6
  C.l1_xyz = alloc(12288);     // B*512*3
  C.l1_p   = alloc(524288);    // B*512*128
  C.l2_xyz = alloc(3072);      // B*128*3
  C.l2_p   = alloc(262144);    // B*128*256
  C.l3_p   = alloc(8192);      // B*1024
  C.l2_p2  = alloc(262144);    // fp3 out
  C.l1_p2  = alloc(524288);    // fp2 out
  float* f_pick  = alloc(4194304);  // B*4096*128
  float* f_place = alloc(4194304);
  C.fps1 = (int*)alloc(4096);
  C.bq1  = (int*)alloc(131072);
  C.fps2 = (int*)alloc(1024);
  C.bq2  = (int*)alloc(65536);
  C.Xbuf = alloc(8650752);     // max padded rows (sa2: 65536*132)
  C.ping = alloc(16777216);    // max activation (131072*128)
  C.pong = alloc(8388608);     // max intermediate (131072*64)
  float* packArena = alloc(262144);  // zero-padded ragged-K weights (~203k used)

  // ----- pack ragged-K weights (K -> K4 with zero rows) -----
  float* pw = packArena;
  pack_branch(pick, pw, stream);
  pack_branch(place, pw, stream);

  // ----- both branches -----
  run_branch(xyzA, pick,  C, f_pick,  stream);
  run_branch(xyzB, place, C, f_place, stream);

  // ----- head -----
  const long Mh = (long)BATCH * NPTS;
  // mixed = [pick_feat bcast (128), f_place (128)] ; pick_feat = f_pick[:,0,:]
  k_concat<<<nblk(Mh * 256, 256), 256, 0, stream>>>(
      f_place, 128, f_pick, 128, (long)NPTS * 128, NPTS, C.Xbuf, Mh, 256, 1);
  Lyr h1{c1W, c1b, g1, be1, 256, 128};
  launch_linear(C.Xbuf, 256, h1, C.ping, Mh, stream);
  k_head<<<nblk(Mh, 256), 256, 0, stream>>>(C.ping, c2W, c2b, (float*)d_out, Mh, 128);
}